// LatentQuantizer_31885837206161
// MI455X (gfx1250) — compile-verified
//
#include <hip/hip_runtime.h>

// Problem sizes (match reference).
#define B_SZ 2048
#define L_SZ 512
#define K_SZ 512
#define NT   256                 // threads per block (8 waves on wave32)
#define REPS (B_SZ / NT)         // 8 batch elements per thread

// gfx1250 async global->LDS path (ASYNCcnt-tracked), guarded so host pass and
// older toolchains still compile.
#if defined(__AMDGCN__) && __has_builtin(__builtin_amdgcn_global_load_async_to_lds_b64)
#define USE_ASYNC_LDS 1
#else
#define USE_ASYNC_LDS 0
#endif

#if USE_ASYNC_LDS
typedef int v2i __attribute__((ext_vector_type(2)));
typedef __attribute__((address_space(1))) v2i* gptr_v2i;
typedef __attribute__((address_space(3))) v2i* lptr_v2i;
#endif

// One block per codebook row l. Sorts the row in LDS (rank sort with
// index tie-break), then binary-searches each z[b,l] against the sorted row:
// 9 steps instead of a 512-wide scan -> kernel runs near the HBM roofline.
__global__ __launch_bounds__(NT) void vq_quantize_kernel(
    const float* __restrict__ z, const float* __restrict__ cb,
    float* __restrict__ out, float* __restrict__ partials)
{
    __shared__ __align__(16) float raw[K_SZ];   // codebook row, original order
    __shared__ __align__(16) float sval[K_SZ];  // sorted values
    __shared__ int   sidx[K_SZ];                // original index of sorted value
    __shared__ float wavesum[NT / 32];

    const int l = blockIdx.x;
    const int t = threadIdx.x;
    const float* row = cb + (size_t)l * K_SZ;

    // ---- Stage codebook row into LDS (async DMA path on gfx1250) ----
#if USE_ASYNC_LDS
    {
        float* g = const_cast<float*>(row) + 2 * t;   // 8 bytes/lane, 256 lanes = 2KB
        __builtin_amdgcn_global_load_async_to_lds_b64(
            (gptr_v2i)g,
            (lptr_v2i)(&raw[2 * t]),
            0, 0);
#if __has_builtin(__builtin_amdgcn_s_wait_asynccnt)
        __builtin_amdgcn_s_wait_asynccnt(0);
#else
        asm volatile("s_wait_asynccnt 0" ::: "memory");
#endif
    }
#else
    raw[2 * t]     = row[2 * t];
    raw[2 * t + 1] = row[2 * t + 1];
#endif
    __syncthreads();

    // ---- Rank sort (O(K^2) total across block, LDS-broadcast reads) ----
    const int   e0 = t, e1 = t + NT;
    const float v0 = raw[e0], v1 = raw[e1];
    int r0 = 0, r1 = 0;
#pragma unroll 8
    for (int j = 0; j < K_SZ; ++j) {
        const float w = raw[j];
        r0 += (int)((w < v0) || ((w == v0) && (j < e0)));
        r1 += (int)((w < v1) || ((w == v1) && (j < e1)));
    }
    sval[r0] = v0; sidx[r0] = e0;
    sval[r1] = v1; sidx[r1] = e1;
    __syncthreads();

    // ---- Quantize: binary search per (b, l) ----
    float acc = 0.0f;
#pragma unroll
    for (int rep = 0; rep < REPS; ++rep) {
        const int   b  = t + rep * NT;
        const float zb = z[(size_t)b * L_SZ + l];

        int lo = 0, hi = K_SZ;        // lower_bound: first sval >= zb
#pragma unroll
        for (int it = 0; it < 9; ++it) {
            const int  mid = (lo + hi) >> 1;
            const bool go  = sval[mid] < zb;
            lo = go ? mid + 1 : lo;
            hi = go ? hi : mid;
        }

        const int p  = (lo < K_SZ) ? lo : (K_SZ - 1);
        float bv = sval[p];
        int   bi = sidx[p];
        float bd = fabsf(bv - zb);
        if (lo > 0) {
            const float v  = sval[lo - 1];          // strictly < zb
            const float d  = fabsf(v - zb);
            const int   i2 = sidx[lo - 1];
            if (d < bd || (d == bd && i2 < bi)) { bv = v; bi = i2; bd = d; }
        }

        out[(size_t)b * L_SZ + l] = bv;                                    // z_q_st
        out[(size_t)(B_SZ * L_SZ) + 1 + (size_t)b * L_SZ + l] = (float)bi; // indices
        const float diff = bv - zb;
        acc += diff * diff;
    }

    // ---- Deterministic block reduction (wave32 shuffles) ----
#pragma unroll
    for (int off = 16; off > 0; off >>= 1)
        acc += __shfl_xor(acc, off, 32);
    if ((t & 31) == 0) wavesum[t >> 5] = acc;
    __syncthreads();
    if (t == 0) {
        float s = 0.0f;
#pragma unroll
        for (int w = 0; w < NT / 32; ++w) s += wavesum[w];
        partials[l] = s;   // one partial per block; fixed order => deterministic
    }
}

// Fold 512 block partials in a fixed tree; write scalar loss.
__global__ __launch_bounds__(NT) void vq_loss_reduce_kernel(
    const float* __restrict__ partials, float* __restrict__ out)
{
    __shared__ float sm[NT];
    const int t = threadIdx.x;
    sm[t] = partials[t] + partials[t + NT];
    __syncthreads();
#pragma unroll
    for (int off = NT / 2; off > 0; off >>= 1) {
        if (t < off) sm[t] += sm[t + off];
        __syncthreads();
    }
    if (t == 0) {
        // 0.25*mean + 1.0*mean  (both stop_gradient terms equal in fwd value)
        out[(size_t)B_SZ * L_SZ] =
            1.25f * sm[0] * (1.0f / (float)(B_SZ * L_SZ));
    }
}

extern "C" void kernel_launch(void* const* d_in, const int* in_sizes, int n_in,
                              void* d_out, int out_size, void* d_ws, size_t ws_size,
                              hipStream_t stream)
{
    (void)in_sizes; (void)n_in; (void)out_size; (void)ws_size;
    const float* z_batch  = (const float*)d_in[0];   // (B, L) f32
    const float* codebook = (const float*)d_in[1];   // (L, K) f32
    // d_in[2] = iter (unused scalar)

    float* out      = (float*)d_out;   // [B*L z_q | 1 loss | B*L indices-as-float]
    float* partials = (float*)d_ws;    // 512 floats of scratch

    vq_quantize_kernel<<<dim3(L_SZ), dim3(NT), 0, stream>>>(
        z_batch, codebook, out, partials);
    vq_loss_reduce_kernel<<<dim3(1), dim3(NT), 0, stream>>>(partials, out);
}